// TensorNeRF_27994596836217
// MI455X (gfx1250) — compile-verified
//
#include <hip/hip_runtime.h>
#include <hip/hip_bf16.h>
#include <stdint.h>

// ---------------------------------------------------------------------------
// NeRF volume-render compositing, MI455X (gfx1250).
//
// Pure HBM-streaming problem: 65536 rays x 512 samples x 20 B/sample = 671 MB
// read, ~29 us roofline @ 23.3 TB/s, ~1 FLOP/byte -> memory bound, WMMA not
// applicable (batched per-ray contraction + sequential cumprod scan).
//
// Design:
//  - wave32 per ray, 32-sample chunks -> every global request is 32
//    contiguous floats (128 B), perfectly coalesced.
//  - CDNA5 async global->LDS DMA (ASYNCcnt) with a 4-buffer ring and
//    prefetch distance 3 (3 chunks / 1.9 KB in flight per wave) + NT
//    temporal hints (671 MB streamed once through 192 MB L2).
//  - exclusive cumprod via 5-step wave prefix-product (shuffles), running
//    transmittance carried across chunks by lane-31 broadcast.
// ---------------------------------------------------------------------------

#define S_SAMPLES        512
#define CHUNK            32
#define NCHUNK           (S_SAMPLES / CHUNK)   // 16
#define WAVES_PER_BLOCK  8
#define THREADS          (WAVES_PER_BLOCK * 32)
#define NBUF             4                      // async ring depth
#define DSCALE           25.0f

template<int N>
__device__ __forceinline__ void wait_asynccnt() {
    asm volatile("s_wait_asynccnt %0" :: "i"(N) : "memory");
}

// Per-wave staging buffer layout (floats): [0,32) sigma | [32,64) dists |
// [64,160) rgb (contiguous 96 floats of the chunk, sample-major).
__global__ __launch_bounds__(THREADS)
void nerf_composite(const float* __restrict__ sigma,
                    const float* __restrict__ dists,
                    const float* __restrict__ rgb,
                    const float* __restrict__ bg,
                    float* __restrict__ out,
                    int nrays)
{
    __shared__ float lds[WAVES_PER_BLOCK][NBUF][160];

    const int lane = threadIdx.x & 31;
    const int wave = threadIdx.x >> 5;
    const int ray  = blockIdx.x * WAVES_PER_BLOCK + wave;
    if (ray >= nrays) return;

    const float* sigRay = sigma + (size_t)ray * S_SAMPLES;
    const float* dstRay = dists + (size_t)ray * S_SAMPLES;
    const float* rgbRay = rgb   + (size_t)ray * (S_SAMPLES * 3);

    // LDS byte addresses for the async-DMA destination VGPR. Flat addresses
    // in the LDS aperture carry the LDS byte offset in bits [31:0].
    const uint32_t laneB = (uint32_t)lane * 4u;
    uint32_t sA[NBUF], dA[NBUF], rA[NBUF];
    #pragma unroll
    for (int b = 0; b < NBUF; ++b) {
        sA[b] = (uint32_t)(uintptr_t)(&lds[wave][b][0]) + laneB;
        dA[b] = sA[b] + 128u;   // dists region
        rA[b] = sA[b] + 256u;   // rgb region
    }

    // Issue one chunk's DMA: 5x async b32 per lane (sigma, dists, rgb x3).
    // GVS mode: the memory address is the SGPR base (wave-uniform ray/chunk pointer) plus a per-lane
    // VGPR offset; the instruction offset is applied to BOTH the memory and
    // the LDS address, so rgb floats j = lane, lane+32, lane+64 land at
    // float 64 + j of the wave's LDS staging buffer. Tracked on ASYNCcnt; NT so we don't rinse L2.
    auto issue = [&](int c, int buf) {
        const float* sp = sigRay + c * CHUNK;
        const float* dp = dstRay + c * CHUNK;
        const float* rp = rgbRay + c * (CHUNK * 3);
        asm volatile("global_load_async_to_lds_b32 %0, %1, %2 th:TH_LOAD_NT"
                     :: "v"(sA[buf]), "v"(laneB), "s"(sp) : "memory");
        asm volatile("global_load_async_to_lds_b32 %0, %1, %2 th:TH_LOAD_NT"
                     :: "v"(dA[buf]), "v"(laneB), "s"(dp) : "memory");
        asm volatile("global_load_async_to_lds_b32 %0, %1, %2 th:TH_LOAD_NT"
                     :: "v"(rA[buf]), "v"(laneB), "s"(rp) : "memory");
        asm volatile("global_load_async_to_lds_b32 %0, %1, %2 offset:128 th:TH_LOAD_NT"
                     :: "v"(rA[buf]), "v"(laneB), "s"(rp) : "memory");
        asm volatile("global_load_async_to_lds_b32 %0, %1, %2 offset:256 th:TH_LOAD_NT"
                     :: "v"(rA[buf]), "v"(laneB), "s"(rp) : "memory");
    };

    float T = 1.0f;                      // running exclusive transmittance
    float accW = 0.f, accR = 0.f, accG = 0.f, accB = 0.f;

    // Prime the ring: chunks 0..2 in flight.
    issue(0, 0);
    issue(1, 1);
    issue(2, 2);

    for (int c = 0; c < NCHUNK; ++c) {
        const int buf = c & (NBUF - 1);

        // Keep prefetch distance 3; async loads complete in order, so
        // ASYNCcnt <= 5 * (#chunks still wanted in flight) guarantees the
        // chunk we are about to consume has fully landed in LDS.
        if (c + 3 < NCHUNK) {
            issue(c + 3, (c + 3) & (NBUF - 1));
            wait_asynccnt<15>();
        } else if (c == NCHUNK - 3) {
            wait_asynccnt<10>();
        } else if (c == NCHUNK - 2) {
            wait_asynccnt<5>();
        } else {
            wait_asynccnt<0>();
        }

        const float sg = lds[wave][buf][lane];
        const float dt = lds[wave][buf][32 + lane];
        const float r0 = lds[wave][buf][64 + lane * 3 + 0];
        const float g0 = lds[wave][buf][64 + lane * 3 + 1];
        const float b0 = lds[wave][buf][64 + lane * 3 + 2];

        const float a = 1.0f - __expf(-sg * dt * DSCALE);
        const float t = 1.0f - a + 1e-10f;

        // Inclusive prefix product across the wave (5 shuffle steps).
        float p = t;
        #pragma unroll
        for (int i = 1; i < 32; i <<= 1) {
            float q = __shfl_up(p, i, 32);
            if (lane >= i) p *= q;
        }
        // Exclusive product; weight = alpha * T_exclusive.
        float pex = __shfl_up(p, 1, 32);
        if (lane == 0) pex = 1.0f;
        const float w = a * T * pex;

        accW += w;
        accR += w * fminf(fmaxf(r0, 0.0f), 1.0f);
        accG += w * fminf(fmaxf(g0, 0.0f), 1.0f);
        accB += w * fminf(fmaxf(b0, 0.0f), 1.0f);

        // Carry transmittance: total product of this chunk (lane 31).
        T *= __shfl(p, 31, 32);
    }

    // Wave reduction of the four accumulators.
    #pragma unroll
    for (int i = 16; i >= 1; i >>= 1) {
        accW += __shfl_down(accW, i, 32);
        accR += __shfl_down(accR, i, 32);
        accG += __shfl_down(accG, i, 32);
        accB += __shfl_down(accB, i, 32);
    }

    if (lane == 0) {
        const float inv = 1.0f - accW;
        out[ray * 3 + 0] = accR + inv * bg[0];
        out[ray * 3 + 1] = accG + inv * bg[1];
        out[ray * 3 + 2] = accB + inv * bg[2];
    }
}

extern "C" void kernel_launch(void* const* d_in, const int* in_sizes, int n_in,
                              void* d_out, int out_size, void* d_ws, size_t ws_size,
                              hipStream_t stream) {
    const float* sigma = (const float*)d_in[0];   // [B, S]
    const float* dists = (const float*)d_in[1];   // [B, S]
    const float* rgb   = (const float*)d_in[2];   // [B, S, 3]
    const float* bg    = (const float*)d_in[3];   // [3]
    float* out = (float*)d_out;                   // [B, 3]

    const int nrays  = in_sizes[0] / S_SAMPLES;   // B = 65536
    const int blocks = (nrays + WAVES_PER_BLOCK - 1) / WAVES_PER_BLOCK;
    nerf_composite<<<blocks, THREADS, 0, stream>>>(sigma, dists, rgb, bg, out, nrays);
}